// LDSMIXTURELQRRNN_58445914964046
// MI455X (gfx1250) — compile-verified
//
#include <hip/hip_runtime.h>
#include <hip/hip_bf16.h>
#include <math.h>

// ---------------------------------------------------------------------------
// LQR-RNN mixture scan: T=2M step serial recurrence, state = (y[2], h[4]).
// Latency-bound single-wave kernel. CDNA5 paths used:
//   * v_wmma_f32_16x16x4_f32   : h' = Whh*h + n1*c + b  AND  t = s*(Who.h'+bo)
//     in one op via an extended+duplicated A fragment, persistent in-place
//     accumulator (no C->D register copies).
//   * v_tanh_f32 (gfx1250)     : sigmoid(z) = 0.5 + 0.5*tanh(z/2)
//   * global_load_async_to_lds : double-buffered input streaming (s_wait_asynccnt)
//   * global_prefetch_b8       : deep prefetch into GL2
// Critical path uses raw v_sqrt_f32 / v_tanh_f32 / v_med3.
// ---------------------------------------------------------------------------

typedef __attribute__((ext_vector_type(2))) float v2f;
typedef __attribute__((ext_vector_type(8))) float v8f;
typedef int v2i_vs __attribute__((__vector_size__(8)));

#define NEGL2E (-1.44269504088896340736f)

#if defined(__has_builtin)
#if __has_builtin(__builtin_amdgcn_global_load_async_to_lds_b64)
#define USE_ASYNC_BUILTIN 1
#endif
#if __has_builtin(__builtin_amdgcn_s_wait_asynccnt)
#define USE_WAIT_BUILTIN 1
#endif
#if __has_builtin(__builtin_amdgcn_wmma_f32_16x16x4_f32)
#define USE_WMMA 1
#endif
#if __has_builtin(__builtin_amdgcn_sqrtf)
#define FAST_SQRT(x) __builtin_amdgcn_sqrtf(x)
#else
#define FAST_SQRT(x) sqrtf(x)
#endif
// --- gate: t = Z_SCALE * z (folded into WMMA row 4); w = sigmoid(z) --------
#if __has_builtin(__builtin_amdgcn_tanhf)
#define Z_SCALE 0.5f     /* t = z/2 */
#define Z_CLAMP 5.0f     /* clamp(z,±10)/2 */
#define W_FROM_T(tc) fmaf(0.5f, __builtin_amdgcn_tanhf(tc), 0.5f)
#elif __has_builtin(__builtin_amdgcn_exp2f) && __has_builtin(__builtin_amdgcn_rcpf)
#define Z_SCALE NEGL2E   /* t = -z*log2e */
#define Z_CLAMP (14.4269504088896340736f)
#define W_FROM_T(tc) __builtin_amdgcn_rcpf(1.0f + __builtin_amdgcn_exp2f(tc))
#else
#define Z_SCALE NEGL2E
#define Z_CLAMP (14.4269504088896340736f)
#define W_FROM_T(tc) (1.0f / (1.0f + exp2f(tc)))
#endif
#if __has_builtin(__builtin_amdgcn_fmed3f)
#define CLAMP_T(t) __builtin_amdgcn_fmed3f((t), -Z_CLAMP, Z_CLAMP)
#else
#define CLAMP_T(t) fminf(fmaxf((t), -Z_CLAMP), Z_CLAMP)
#endif
#else
#define FAST_SQRT(x) sqrtf(x)
#define Z_SCALE NEGL2E
#define Z_CLAMP (14.4269504088896340736f)
#define W_FROM_T(tc) (1.0f / (1.0f + exp2f(tc)))
#define CLAMP_T(t) fminf(fmaxf((t), -Z_CLAMP), Z_CLAMP)
#endif

__device__ __forceinline__ void async_ld_b64(const float* g, const float* l) {
#ifdef USE_ASYNC_BUILTIN
  __builtin_amdgcn_global_load_async_to_lds_b64(
      (__attribute__((address_space(1))) v2i_vs*)(unsigned long long)g,
      (__attribute__((address_space(3))) v2i_vs*)(unsigned)(unsigned long long)l,
      0, 0);
#else
  unsigned ldsa = (unsigned)(unsigned long long)l;
  unsigned long long ga = (unsigned long long)g;
  asm volatile("global_load_async_to_lds_b64 %0, %1, off"
               :: "v"(ldsa), "v"(ga) : "memory");
#endif
}

__device__ __forceinline__ void wait_async2() {
#ifdef USE_WAIT_BUILTIN
  __builtin_amdgcn_s_wait_asynccnt(2);
#else
  asm volatile("s_wait_asynccnt 2" ::: "memory");
#endif
  asm volatile("" ::: "memory");
}

__device__ __forceinline__ void wait_async0() {
#ifdef USE_WAIT_BUILTIN
  __builtin_amdgcn_s_wait_asynccnt(0);
#else
  asm volatile("s_wait_asynccnt 0" ::: "memory");
#endif
  asm volatile("" ::: "memory");
}

__global__ __launch_bounds__(32) void lqr_scan_kernel(
    const float* __restrict__ xs1, const float* __restrict__ xs2,
    const float* __restrict__ y0p, const float* __restrict__ g1p,
    const float* __restrict__ g2p, const float* __restrict__ Wxh,
    const float* __restrict__ Whh, const float* __restrict__ bh,
    const float* __restrict__ h0p, const float* __restrict__ Who,
    const float* __restrict__ bop, float* __restrict__ out, int T) {
  __shared__ __align__(16) float sb[2][128];  // [phase][ xs1:0..63 | xs2:64..127 ]
  const int lane = threadIdx.x;
  const int CH = 32;
  const int NC = T / CH;

  float* __restrict__ xout = out;                   // (T,)
  float* __restrict__ yout = out + T;               // (T+1, 2)
  float* __restrict__ uout = out + T + 2 * (T + 1); // (T, 2)

  // ---- uniform constants ------------------------------------------------
  const float g1 = g1p[0], g2 = g2p[0], bo = bop[0];
  const float BP = 1.0f / 60.0f;
  const float who0 = Who[0], who1 = Who[1], who2 = Who[2], who3 = Who[3];
  const float bh0 = bh[0], bh1 = bh[1], bh2 = bh[2], bh3 = bh[3];
  // W_xh.T @ [n1,n1] = n1 * (row0 + row1)
  const float ce0 = Wxh[0] + Wxh[4];
  const float ce1 = Wxh[1] + Wxh[5];
  const float ce2 = Wxh[2] + Wxh[6];
  const float ce3 = Wxh[3] + Wxh[7];
  // row-4 terms prescaled by Z_SCALE: WMMA directly yields t = Z_SCALE*z
  const float ce4s = (who0 * ce0 + who1 * ce1 + who2 * ce2 + who3 * ce3) * Z_SCALE;
  const float be4s = (who0 * bh0 + who1 * bh1 + who2 * bh2 + who3 * bh3 + bo) * Z_SCALE;

  const float w00 = Whh[0],  w01 = Whh[1],  w02 = Whh[2],  w03 = Whh[3];
  const float w10 = Whh[4],  w11 = Whh[5],  w12 = Whh[6],  w13 = Whh[7];
  const float w20 = Whh[8],  w21 = Whh[9],  w22 = Whh[10], w23 = Whh[11];
  const float w30 = Whh[12], w31 = Whh[13], w32 = Whh[14], w33 = Whh[15];

#ifdef USE_WMMA
  // ---- constant A fragment ----------------------------------------------
  // rows 0-3: Whh ; row 4: Z_SCALE * (Who^T Whh) ; rows 8-12 duplicate 0-4
  // 32-bit A 16x4 layout: lane -> M = lane&15, K pair = (lane&16)?{2,3}:{0,1}
  const float wz0 = (who0 * w00 + who1 * w10 + who2 * w20 + who3 * w30) * Z_SCALE;
  const float wz1 = (who0 * w01 + who1 * w11 + who2 * w21 + who3 * w31) * Z_SCALE;
  const float wz2 = (who0 * w02 + who1 * w12 + who2 * w22 + who3 * w32) * Z_SCALE;
  const float wz3 = (who0 * w03 + who1 * w13 + who2 * w23 + who3 * w33) * Z_SCALE;
  const int me = lane & 15;
  const bool hiK = (lane & 16) != 0;
  const float a0r0 = hiK ? w02 : w00, a1r0 = hiK ? w03 : w01;
  const float a0r1 = hiK ? w12 : w10, a1r1 = hiK ? w13 : w11;
  const float a0r2 = hiK ? w22 : w20, a1r2 = hiK ? w23 : w21;
  const float a0r3 = hiK ? w32 : w30, a1r3 = hiK ? w33 : w31;
  const float a0r4 = hiK ? wz2 : wz0, a1r4 = hiK ? wz3 : wz1;
  const int r = (me < 5) ? me : ((me >= 8 && me < 13) ? (me - 8) : -1);
  const float A0 = (r == 0) ? a0r0 : (r == 1) ? a0r1 : (r == 2) ? a0r2
                 : (r == 3) ? a0r3 : (r == 4) ? a0r4 : 0.0f;
  const float A1 = (r == 0) ? a1r0 : (r == 1) ? a1r1 : (r == 2) ? a1r2
                 : (r == 3) ? a1r3 : (r == 4) ? a1r4 : 0.0f;
  v2f afrag;
  afrag.x = A0;
  afrag.y = A1;
  const bool lo16 = (lane < 16);
  // persistent accumulator: [h0,h1,h2,h3,t, 0,0,0]; rows 5-7 stay 0 forever
  // (A rows 5-7/13-15 are zero so D[5..7] = C[5..7]); updated IN PLACE.
  v8f acc;
  acc[0] = h0p[0]; acc[1] = h0p[1]; acc[2] = h0p[2]; acc[3] = h0p[3];
  acc[4] = 0.f; acc[5] = 0.f; acc[6] = 0.f; acc[7] = 0.f;
#else
  float h0v = h0p[0], h1v = h0p[1], h2v = h0p[2], h3v = h0p[3];
#endif

  // ---- state ------------------------------------------------------------
  float yx = y0p[0], yy = y0p[1];
  if (lane == 0) { yout[0] = yx; yout[1] = yy; }

  // per-lane output staging (lane l keeps step t0+l of each chunk)
  float wacc = 0.f, yxa = 0.f, yya = 0.f, uxa = 0.f, uya = 0.f;

  // ---- prime the async double buffer (chunk 0 -> phase 0) ---------------
  async_ld_b64(xs1 + 2 * lane, &sb[0][2 * lane]);
  async_ld_b64(xs2 + 2 * lane, &sb[0][64 + 2 * lane]);

  for (int k = 0; k < NC; ++k) {
    const int p = k & 1;
    if (k + 1 < NC) {
      const int kn = k + 1;
      async_ld_b64(xs1 + kn * 64 + 2 * lane, &sb[kn & 1][2 * lane]);
      async_ld_b64(xs2 + kn * 64 + 2 * lane, &sb[kn & 1][64 + 2 * lane]);
      if (k + 9 < NC) {  // deep prefetch into GL2
        __builtin_prefetch(xs1 + (k + 9) * 64 + 2 * lane, 0, 0);
        __builtin_prefetch(xs2 + (k + 9) * 64 + 2 * lane, 0, 0);
      }
      wait_async2();  // chunk k's 2 loads complete; k+1's may be in flight
    } else {
      wait_async0();
    }

    const float* sp = &sb[p][0];
#pragma unroll
    for (int i = 0; i < CH; ++i) {
      const float x1x = sp[2 * i], x1y = sp[2 * i + 1];
      const float x2x = sp[64 + 2 * i], x2y = sp[64 + 2 * i + 1];
      const float e1x = yx - x1x, e1y = yy - x1y;
      const float e2x = yx - x2x, e2y = yy - x2y;
      const float n1 = FAST_SQRT(fmaf(e1x, e1x, e1y * e1y));

      // LQR mix terms that do not depend on w: compute during tanh latency
      const float px = g2 * e2x, py = g2 * e2y;          //  g2*e2
      const float qx = fmaf(-g1, e1x, px);               //  g2*e2 - g1*e1
      const float qy = fmaf(-g1, e1y, py);

      float t;  // Z_SCALE * z, pre-clamp
#ifdef USE_WMMA
      v2f bb;  // B 4x16: every column = h_prev  (K = vgpr + 2*lanehalf)
      bb.x = lo16 ? acc[0] : acc[2];
      bb.y = lo16 ? acc[1] : acc[3];
      // overwrite C rows 0-4 in place; rows 5-7 untouched (stay 0)
      acc[0] = fmaf(n1, ce0, bh0);
      acc[1] = fmaf(n1, ce1, bh1);
      acc[2] = fmaf(n1, ce2, bh2);
      acc[3] = fmaf(n1, ce3, bh3);
      acc[4] = fmaf(n1, ce4s, be4s);
      acc = __builtin_amdgcn_wmma_f32_16x16x4_f32(
          false, afrag, false, bb, (short)0, acc, false, false);
      t = acc[4];
#else
      const float l0 = w00 * h0v + w01 * h1v + w02 * h2v + w03 * h3v;
      const float l1 = w10 * h0v + w11 * h1v + w12 * h2v + w13 * h3v;
      const float l2 = w20 * h0v + w21 * h1v + w22 * h2v + w23 * h3v;
      const float l3 = w30 * h0v + w31 * h1v + w32 * h2v + w33 * h3v;
      h0v = fmaf(n1, ce0, bh0) + l0;
      h1v = fmaf(n1, ce1, bh1) + l1;
      h2v = fmaf(n1, ce2, bh2) + l2;
      h3v = fmaf(n1, ce3, bh3) + l3;
      const float zz = who0 * h0v + who1 * h1v + who2 * h2v + who3 * h3v + bo;
      t = zz * Z_SCALE;
#endif
      // clamp in scaled domain (monotone scale => interval clamp commutes)
      const float tc = CLAMP_T(t);
      const float w = W_FROM_T(tc);  // sigmoid(clamp(z,-10,10))

      // u = w*(g2*e2 - g1*e1) - g2*e2   (one FMA after w resolves)
      const float ux = fmaf(w, qx, -px);
      const float uy = fmaf(w, qy, -py);
      yx = fmaf(BP, ux, yx);
      yy = fmaf(BP, uy, yy);

      const bool keep = (lane == i);
      wacc = keep ? w : wacc;
      yxa = keep ? yx : yxa;
      yya = keep ? yy : yya;
      uxa = keep ? ux : uxa;
      uya = keep ? uy : uya;
    }

    // coalesced chunk stores (off the serial critical path)
    const int t0 = k * CH;
    xout[t0 + lane] = wacc;
    float2 yv; yv.x = yxa; yv.y = yya;
    *(float2*)(yout + 2 * (t0 + 1 + lane)) = yv;
    float2 uv; uv.x = uxa; uv.y = uya;
    *(float2*)(uout + 2 * (t0 + lane)) = uv;
  }
}

extern "C" void kernel_launch(void* const* d_in, const int* in_sizes, int n_in,
                              void* d_out, int out_size, void* d_ws, size_t ws_size,
                              hipStream_t stream) {
  (void)n_in; (void)out_size; (void)d_ws; (void)ws_size;
  const int T = in_sizes[0] / 2;  // xstar1 is (T,2)
  lqr_scan_kernel<<<dim3(1), dim3(32), 0, stream>>>(
      (const float*)d_in[0],  // xstar1
      (const float*)d_in[1],  // xstar2
      (const float*)d_in[2],  // init_y
      (const float*)d_in[3],  // gain1
      (const float*)d_in[4],  // gain2
      (const float*)d_in[5],  // W_xh (2,4)
      (const float*)d_in[6],  // W_hh (4,4)
      (const float*)d_in[7],  // b_h  (4,)
      (const float*)d_in[8],  // h0   (4,)
      (const float*)d_in[9],  // W_ho (4,1)
      (const float*)d_in[10], // b_o  (1,)
      (float*)d_out, T);
}